// LinearDecoderUZ_15582141350496
// MI455X (gfx1250) — compile-verified
//
#include <hip/hip_runtime.h>

typedef __attribute__((ext_vector_type(2))) float v2f;
typedef __attribute__((ext_vector_type(4))) float f32x4;
typedef __attribute__((ext_vector_type(8))) float v8f;

#define NL 128           // n_latent == n_out == 128
#define TILE 16          // samples per chunk (WMMA M)
#define LDS_STRIDE 132   // 128 + 4 pad to break LDS bank conflicts

// ---------------- binning pre-pass ----------------

__global__ void k_zero(int* p, int n) {
    int i = blockIdx.x * blockDim.x + threadIdx.x;
    if (i < n) p[i] = 0;
}

__global__ void k_hist(const int* __restrict__ donor_id, int* __restrict__ counts, int B) {
    int i = blockIdx.x * blockDim.x + threadIdx.x;
    if (i < B) atomicAdd(&counts[donor_id[i]], 1);
}

// single-block Hillis-Steele scan; blockDim.x must be >= n (n_donors=500)
__global__ void k_scan(const int* __restrict__ counts, int* __restrict__ starts,
                       int* __restrict__ cursor, int n) {
    __shared__ int sm[1024];
    int t = threadIdx.x;
    int my = (t < n) ? counts[t] : 0;
    sm[t] = my;
    __syncthreads();
    for (int off = 1; off < (int)blockDim.x; off <<= 1) {
        int v = (t >= off) ? sm[t - off] : 0;
        __syncthreads();
        sm[t] += v;
        __syncthreads();
    }
    if (t < n) {
        int excl = sm[t] - my;
        starts[t] = excl;
        cursor[t] = excl;
        if (t == n - 1) starts[n] = sm[t];
    }
}

__global__ void k_scatter(const int* __restrict__ donor_id, int* __restrict__ cursor,
                          int* __restrict__ sample_of, int B) {
    int i = blockIdx.x * blockDim.x + threadIdx.x;
    if (i < B) {
        int d = donor_id[i];
        int pos = atomicAdd(&cursor[d], 1);
        sample_of[pos] = i;
    }
}

// ---------------- grouped-GEMM main kernel ----------------
// grid.x = n_donors; block = 256 threads = 8 waves; wave w owns output cols [16w, 16w+16)

__global__ __launch_bounds__(256)
void k_decoder_wmma(const float* __restrict__ u,
                    const float* __restrict__ amat,
                    const float* __restrict__ offs,
                    const int*   __restrict__ starts,
                    const int*   __restrict__ sample_of,
                    float*       __restrict__ out) {
    __shared__ float a_tile[TILE * LDS_STRIDE];  // u rows for this chunk (zero-padded)
    __shared__ int   row_sample[TILE];           // global sample index per row (-1 = pad)

    const int d    = blockIdx.x;
    const int t    = threadIdx.x;
    const int lane = t & 31;
    const int wave = t >> 5;

    const int s0 = starts[d];
    const int nd = starts[d + 1] - s0;
    if (nd == 0) return;  // uniform per block

    // --- preload this wave's 128x16 B panel into registers ---
    // B frag layout for V_WMMA_F32_16X16X4_F32 (4x16, KxN):
    //   v0: lanes 0-15 -> K=k0,   N=lane ; lanes 16-31 -> K=k0+2, N=lane-16
    //   v1: lanes 0-15 -> K=k0+1          ; lanes 16-31 -> K=k0+3
    const int kb_lane = (lane >> 4) << 1;        // 0 or 2
    const int col     = (wave << 4) + (lane & 15);  // global output column 0..127
    const float* Bp   = amat + (size_t)d * NL * NL;
    v2f bfrag[32];
    #pragma unroll
    for (int ks = 0; ks < 32; ++ks) {
        int k = (ks << 2) + kb_lane;
        bfrag[ks].x = Bp[(size_t)k * NL + col];
        bfrag[ks].y = Bp[(size_t)(k + 1) * NL + col];
    }
    const float off_val = offs[(size_t)d * NL + col];

    const int nchunks = (nd + TILE - 1) / TILE;
    for (int c = 0; c < nchunks; ++c) {
        __syncthreads();  // previous chunk's LDS reads done before overwrite

        // cooperatively stage 16 u-rows into LDS: 256 threads * 8 floats
        {
            int row    = t >> 4;          // 0..15
            int cstart = (t & 15) << 3;   // 0,8,...,120
            int m      = c * TILE + row;
            float* dst = &a_tile[row * LDS_STRIDE + cstart];
            if (m < nd) {
                int b = sample_of[s0 + m];
                if ((t & 15) == 0) row_sample[row] = b;
                const float* up = u + (size_t)b * NL + cstart;
                *(f32x4*)dst       = *(const f32x4*)up;
                *(f32x4*)(dst + 4) = *(const f32x4*)(up + 4);
            } else {
                if ((t & 15) == 0) row_sample[row] = -1;
                *(f32x4*)dst       = (f32x4){0.f, 0.f, 0.f, 0.f};
                *(f32x4*)(dst + 4) = (f32x4){0.f, 0.f, 0.f, 0.f};
            }
        }
        __syncthreads();

        // --- accumulate 16x16 tile over K=128 with f32 WMMA ---
        // A frag layout (16x4, MxK): v0: K=k0 (lanes 0-15) / K=k0+2 (lanes 16-31); v1: +1
        v8f acc = {};
        const int arow = lane & 15;
        #pragma unroll
        for (int ks = 0; ks < 32; ++ks) {
            int k = (ks << 2) + kb_lane;
            v2f a;
            a.x = a_tile[arow * LDS_STRIDE + k];
            a.y = a_tile[arow * LDS_STRIDE + k + 1];
            acc = __builtin_amdgcn_wmma_f32_16x16x4_f32(
                /*neg_a=*/false, a, /*neg_b=*/false, bfrag[ks],
                /*c_mod=*/(short)0, acc, /*reuse_a=*/false, /*reuse_b=*/false);
        }

        // --- fused epilogue: out = wmma + u + offsets ---
        // C/D layout: VGPR r -> M = r + 8*(lane/16), N = lane%16
        #pragma unroll
        for (int r = 0; r < 8; ++r) {
            int m   = r + ((lane >> 4) << 3);
            int bsm = row_sample[m];
            if (bsm >= 0) {
                float uval = a_tile[m * LDS_STRIDE + col];  // n_latent == n_out
                out[(size_t)bsm * NL + col] = acc[r] + uval + off_val;
            }
        }
    }
}

// ---------------- launch ----------------

extern "C" void kernel_launch(void* const* d_in, const int* in_sizes, int n_in,
                              void* d_out, int out_size, void* d_ws, size_t ws_size,
                              hipStream_t stream) {
    const float* u        = (const float*)d_in[0];
    const int*   donor_id = (const int*)  d_in[1];
    const float* amat     = (const float*)d_in[2];
    const float* offs     = (const float*)d_in[3];
    float*       out      = (float*)d_out;

    const int B        = in_sizes[1];
    const int n_donors = in_sizes[3] / NL;

    // workspace layout (ints): counts | starts(n+1) | cursor | sample_of(B)
    int* counts    = (int*)d_ws;
    int* starts    = counts + n_donors;
    int* cursor    = starts + n_donors + 1;
    int* sample_of = cursor + n_donors;

    k_zero   <<<(n_donors + 255) / 256, 256, 0, stream>>>(counts, n_donors);
    k_hist   <<<(B + 255) / 256,        256, 0, stream>>>(donor_id, counts, B);
    k_scan   <<<1, 1024,                     0, stream>>>(counts, starts, cursor, n_donors);
    k_scatter<<<(B + 255) / 256,        256, 0, stream>>>(donor_id, cursor, sample_of, B);

    k_decoder_wmma<<<n_donors, 256, 0, stream>>>(u, amat, offs, starts, sample_of, out);
}